// GATTransformerODWeek_43198781063696
// MI455X (gfx1250) — compile-verified
//
#include <hip/hip_runtime.h>
#include <math.h>

typedef float v2f __attribute__((ext_vector_type(2)));
typedef float v8f __attribute__((ext_vector_type(8)));

// ---------------- model constants ----------------
static const int kN   = 1200;
static const int kB   = 4;
static const int kT   = 12;
static const int kFut = 6;
static const int kBN  = kB * kN;          // 4800
static const int kBTN = kB * kT * kN;     // 57600
static const int kEst = 16 * kB * kN;     // 76800 static edges
static const int kEdy = 10 * kB * kT * kN;// 576000 dynamic edges

// ---------------- workspace layout (float offsets, all even) ----------------
static const size_t SZ_BIG    = (size_t)kBTN * 128;                    // 7,372,800
static const size_t OFF_X     = 0;
static const size_t OFF_Y     = SZ_BIG;
static const size_t OFF_XL    = 2 * SZ_BIG;
static const size_t OFF_XR    = 3 * SZ_BIG;
static const size_t OFF_E     = 4 * SZ_BIG;                            // (Edy+BTN)*4 heads
static const size_t OFF_MAXB  = OFF_E    + (size_t)(kEdy + kBTN) * 4;
static const size_t OFF_DEN   = OFF_MAXB + (size_t)kBTN * 4;
static const size_t OFF_FEAT  = OFF_DEN  + (size_t)kBTN * 4;           // 4800*24
static const size_t OFF_SHORT = OFF_FEAT + (size_t)kBN * 24;
static const size_t OFF_SPAT  = OFF_SHORT+ (size_t)kBN * 32;
static const size_t OFF_TENC  = OFF_SPAT + (size_t)kBN * 32;           // 48*32
static const size_t OFF_MEAN  = OFF_TENC + (size_t)kB * kT * 32;       // 1 float (+pad)
static const size_t OFF_FUSED = OFF_MEAN + 4;                          // 12*4800*32
static const size_t WS_NEED   = OFF_FUSED + (size_t)kT * kBN * 32;     // ~34.75M floats

// decoder overlays (GAT scratch regions are dead by then; FUSED stays live)
static const size_t OFF_DECX = OFF_X;                 //  6*4800*32
static const size_t OFF_Q    = OFF_X + 1048576;       //  6*4800*32
static const size_t OFF_K    = OFF_X + 2097152;       // 12*4800*32
static const size_t OFF_V    = OFF_X + 4194304;       // 12*4800*32
static const size_t OFF_ATT  = OFF_X + 6291456;       //  6*4800*32
static const size_t OFF_FFH  = OFF_Y;                 // 28800*128
static const size_t OFF_PROJ = OFF_Y + 4194304;       //  6*4800*32

// =================================================================
//  WMMA f32 GEMM:  C[M,N] = act( A[M,K] @ op(B) + bias )
//  REQUIRES: M%16==0, N%16==0, K%4==0 (all call sites satisfy this).
//  op(B) = B[K,N] row-major (transB=0)  or  B[N,K]^T (transB=1)
//  One wave32 computes one 16x16 f32 tile via V_WMMA_F32_16X16X4_F32.
//  No lane-divergent control flow: EXEC is all-ones around every WMMA.
// =================================================================
__global__ __launch_bounds__(256)
void k_wmma_gemm(const float* __restrict__ A, const float* __restrict__ B,
                 const float* __restrict__ bias, float* __restrict__ C,
                 int M, int N, int K, int lda, int ldb, int ldc,
                 int transB, int act)
{
    const int lane   = threadIdx.x & 31;
    const int wave   = threadIdx.x >> 5;
    const int tilesN = N >> 4;
    const int tilesM = M >> 4;
    const int tile   = blockIdx.x * (blockDim.x >> 5) + wave;   // wave-uniform
    if (tile >= tilesM * tilesN) return;
    const int tm = tile / tilesN;
    const int tn = tile - tm * tilesN;
    const int half = lane >> 4;      // K-pair select: {k,k+1} vs {k+2,k+3}
    const int l16  = lane & 15;
    const int m0 = tm << 4, n0 = tn << 4;

    v8f acc = {0.f, 0.f, 0.f, 0.f, 0.f, 0.f, 0.f, 0.f};

    // A tile: lane = row m0+l16, consecutive K pair -> one b64 load per step
    const float* ap = A + (size_t)(m0 + l16) * lda + 2 * half;

    if (!transB) {
        // W[K,N]: column n0+l16, rows k+2*half and k+2*half+1 (two b32 loads)
        const float* bp = B + (size_t)(2 * half) * ldb + (n0 + l16);
        for (int k = 0; k < K; k += 4) {
            v2f a = *(const v2f*)(ap + k);
            v2f b;
            b.x = bp[(size_t)k * ldb];
            b.y = bp[(size_t)(k + 1) * ldb];
            acc = __builtin_amdgcn_wmma_f32_16x16x4_f32(false, a, false, b,
                                                        (short)0, acc, false, false);
        }
    } else {
        // B[N,K]^T: lane = row n0+l16 of B, consecutive K pair -> b64 load
        const float* bp = B + (size_t)(n0 + l16) * ldb + 2 * half;
        for (int k = 0; k < K; k += 4) {
            v2f a = *(const v2f*)(ap + k);
            v2f b = *(const v2f*)(bp + k);
            acc = __builtin_amdgcn_wmma_f32_16x16x4_f32(false, a, false, b,
                                                        (short)0, acc, false, false);
        }
    }

    const int n = n0 + l16;
    const float bv = bias ? bias[n] : 0.f;
    float* cp = C + (size_t)(m0 + half * 8) * ldc + n;   // documented C/D layout
    #pragma unroll
    for (int r = 0; r < 8; ++r) {
        float v = acc[r] + bv;
        if (act == 1)      v = v > 0.f ? v : 0.f;        // relu
        else if (act == 2) v = v > 0.f ? v : expm1f(v);  // elu
        cp[(size_t)r * ldc] = v;
    }
}

// Naive tiny GEMM for the K=2 time-encoding projection (48x32 output).
__global__ void k_small_gemm(const float* __restrict__ A, const float* __restrict__ B,
                             const float* __restrict__ bias, float* __restrict__ C,
                             int M, int N, int K) {
    int i = blockIdx.x * blockDim.x + threadIdx.x;
    if (i >= M * N) return;
    int m = i / N, n = i % N;
    float s = bias[n];
    for (int k = 0; k < K; ++k) s += A[(size_t)m * K + k] * B[(size_t)k * N + n];
    C[i] = s;
}

// ---------------- elementwise / utility kernels ----------------
__global__ void k_fill(float* __restrict__ p, float v, long long n) {
    long long i = (long long)blockIdx.x * blockDim.x + threadIdx.x;
    if (i < n) p[i] = v;
}

__global__ void k_add(const float* __restrict__ a, const float* __restrict__ b,
                      float* __restrict__ c, long long n) {
    long long i = (long long)blockIdx.x * blockDim.x + threadIdx.x;
    if (i < n) c[i] = a[i] + b[i];
}

__global__ void k_build_feat(const float* __restrict__ nemb, const float* __restrict__ wemb,
                             const int* __restrict__ wk, float* __restrict__ x0) {
    long long i = (long long)blockIdx.x * blockDim.x + threadIdx.x;
    const long long tot = (long long)kBN * 24;
    if (i >= tot) return;
    int c = (int)(i % 24);
    int row = (int)(i / 24);
    int b = row / kN, nn = row % kN;
    x0[i] = (c < 16) ? nemb[nn * 16 + c] : wemb[wk[b] * 8 + (c - 16)];
}

__global__ void k_broadcast_dyn(const float* __restrict__ spat, float* __restrict__ out) {
    long long i = (long long)blockIdx.x * blockDim.x + threadIdx.x;
    const long long tot = (long long)kBTN * 32;
    if (i >= tot) return;
    int c = (int)(i & 31);
    long long row = i >> 5;                 // (b*T + t)*N + n
    int nn = (int)(row % kN);
    int b  = (int)(row / ((long long)kT * kN));
    out[i] = spat[((size_t)b * kN + nn) * 32 + c];
}

// fused[t, b*N+n, c] = dyn[b,t,n,c] + spat[b,n,c] + tenc[b,t,c] + pe[t,c] + posl[t,c]
__global__ void k_fused_build(const float* __restrict__ dyn, const float* __restrict__ spat,
                              const float* __restrict__ tenc, const float* __restrict__ posl,
                              float* __restrict__ fused) {
    long long i = (long long)blockIdx.x * blockDim.x + threadIdx.x;
    const long long tot = (long long)kT * kBN * 32;
    if (i >= tot) return;
    int c = (int)(i & 31);
    long long row = i >> 5;                 // t*B*N + b*N + n
    int nn = (int)(row % kN);
    long long r2 = row / kN;
    int b = (int)(r2 % kB);
    int t = (int)(r2 / kB);
    float dv = dyn[(((size_t)b * kT + t) * kN + nn) * 32 + c];
    float sv = spat[((size_t)b * kN + nn) * 32 + c];
    float tv = tenc[((size_t)b * kT + t) * 32 + c];
    int ii = c & ~1;
    float div = __expf((float)ii * (-0.28782313662425572f));  // -ln(10000)/32
    float arg = (float)t * div;
    float pe = (c & 1) ? __cosf(arg) : __sinf(arg);
    fused[i] = dv + sv + tv + pe + posl[(size_t)t * 32 + c];
}

// ---------------- GATv2 kernels (3-pass segment softmax) ----------------
__global__ void k_gat_edge_score(const int* __restrict__ esrc, const int* __restrict__ edst,
                                 int E, int n,
                                 const float* __restrict__ xl, const float* __restrict__ xr,
                                 const float* __restrict__ att, const float* __restrict__ We,
                                 const float* __restrict__ ea, const float* __restrict__ eamean,
                                 float* __restrict__ esc, unsigned* __restrict__ maxbits,
                                 int heads, int outd) {
    long long idx = (long long)blockIdx.x * blockDim.x + threadIdx.x;
    const long long tot = (long long)(E + n) * heads;
    if (idx >= tot) return;
    int e = (int)(idx / heads);
    int h = (int)(idx - (long long)e * heads);
    int src, dst; float eav = 0.f;
    if (e < E) { src = esrc[e]; dst = edst[e]; if (ea) eav = ea[e]; }
    else       { src = e - E; dst = src;       if (ea) eav = *eamean; }
    const int hc = heads * outd;
    const float* pl = xl + (size_t)src * hc + h * outd;
    const float* pr = xr + (size_t)dst * hc + h * outd;
    const float* pa = att + h * outd;
    const float* pw = We ? (We + h * outd) : (const float*)0;
    float s = 0.f;
    for (int c = 0; c < outd; ++c) {
        float m = pl[c] + pr[c];
        if (pw) m += eav * pw[c];
        m = m > 0.f ? m : 0.2f * m;           // leaky_relu(0.2)
        s += m * pa[c];
    }
    esc[idx] = s;
    unsigned bits = __float_as_uint(s);
    bits = (bits & 0x80000000u) ? ~bits : (bits | 0x80000000u);   // order-preserving map
    atomicMax(&maxbits[(size_t)dst * heads + h], bits);
}

__global__ void k_gat_finalize_max(unsigned* __restrict__ mb, float* __restrict__ mf, int tot) {
    int i = blockIdx.x * blockDim.x + threadIdx.x;
    if (i >= tot) return;
    unsigned u = mb[i];
    float f = 0.f;
    if (u != 0u) {
        unsigned b = (u & 0x80000000u) ? (u & 0x7FFFFFFFu) : ~u;
        f = __uint_as_float(b);
        if (!__builtin_isfinite(f)) f = 0.f;
    }
    mf[i] = f;
}

__global__ void k_gat_expsum(const int* __restrict__ edst, int E, int n,
                             float* __restrict__ esc, const float* __restrict__ mf,
                             float* __restrict__ den, int heads) {
    long long idx = (long long)blockIdx.x * blockDim.x + threadIdx.x;
    const long long tot = (long long)(E + n) * heads;
    if (idx >= tot) return;
    int e = (int)(idx / heads);
    int h = (int)(idx - (long long)e * heads);
    int dst = (e < E) ? edst[e] : (e - E);
    float ex = __expf(esc[idx] - mf[(size_t)dst * heads + h]);
    esc[idx] = ex;
    atomicAdd(&den[(size_t)dst * heads + h], ex);
}

__global__ void k_gat_scatter(const int* __restrict__ esrc, const int* __restrict__ edst,
                              int E, int n,
                              const float* __restrict__ esc, const float* __restrict__ den,
                              const float* __restrict__ xl, float* __restrict__ out,
                              int heads, int outd) {
    long long idx = (long long)blockIdx.x * blockDim.x + threadIdx.x;
    const long long tot = (long long)(E + n) * heads;
    if (idx >= tot) return;
    int e = (int)(idx / heads);
    int h = (int)(idx - (long long)e * heads);
    int src, dst;
    if (e < E) { src = esrc[e]; dst = edst[e]; } else { src = dst = e - E; }
    float alpha = esc[idx] / (den[(size_t)dst * heads + h] + 1e-16f);
    const int hc = heads * outd;
    const float* pl = xl + (size_t)src * hc + h * outd;
    float* po = out + (size_t)dst * hc + h * outd;
    for (int c = 0; c < outd; ++c) atomicAdd(&po[c], alpha * pl[c]);
}

__global__ void k_bias_act(float* __restrict__ x, const float* __restrict__ bias,
                           long long rows, int C, int act) {
    long long i = (long long)blockIdx.x * blockDim.x + threadIdx.x;
    if (i >= rows * C) return;
    int c = (int)(i % C);
    float v = x[i] + bias[c];
    if (act == 2) v = v > 0.f ? v : expm1f(v);    // elu
    x[i] = v;
}

// LDS tree reduction for edge-attr mean
__global__ void k_reduce_sum(const float* __restrict__ x, long long n, float* __restrict__ out) {
    __shared__ float sm[256];
    float acc = 0.f;
    for (long long i = (long long)blockIdx.x * blockDim.x + threadIdx.x; i < n;
         i += (long long)gridDim.x * blockDim.x)
        acc += x[i];
    sm[threadIdx.x] = acc;
    __syncthreads();
    for (int s = 128; s > 0; s >>= 1) {
        if ((int)threadIdx.x < s) sm[threadIdx.x] += sm[threadIdx.x + s];
        __syncthreads();
    }
    if (threadIdx.x == 0) atomicAdd(out, sm[0]);
}

__global__ void k_scale1(float* p, float s) {
    if (blockIdx.x == 0 && threadIdx.x == 0) p[0] *= s;
}

// ---------------- decoder kernels ----------------
// d=32, nh=8, hd=4 fixed.  thread per (s, bn, h)
__global__ void k_mha_attn(const float* __restrict__ q, const float* __restrict__ k,
                           const float* __restrict__ v, float* __restrict__ out,
                           int S, int Tk, int Bn) {
    long long idx = (long long)blockIdx.x * blockDim.x + threadIdx.x;
    const long long tot = (long long)S * Bn * 8;
    if (idx >= tot) return;
    int h  = (int)(idx % 8);
    int bn = (int)((idx / 8) % Bn);
    int s  = (int)(idx / ((long long)8 * Bn));
    const float* qp = q + ((size_t)s * Bn + bn) * 32 + h * 4;
    float sc[12];
    float mx = -1e30f;
    for (int t = 0; t < Tk; ++t) {
        const float* kp = k + ((size_t)t * Bn + bn) * 32 + h * 4;
        float d = qp[0]*kp[0] + qp[1]*kp[1] + qp[2]*kp[2] + qp[3]*kp[3];
        d *= 0.5f;                                  // 1/sqrt(hd), hd=4
        sc[t] = d;
        mx = fmaxf(mx, d);
    }
    float den = 0.f;
    for (int t = 0; t < Tk; ++t) { sc[t] = __expf(sc[t] - mx); den += sc[t]; }
    float inv = 1.f / den;
    float o0 = 0.f, o1 = 0.f, o2 = 0.f, o3 = 0.f;
    for (int t = 0; t < Tk; ++t) {
        const float* vp = v + ((size_t)t * Bn + bn) * 32 + h * 4;
        float a = sc[t] * inv;
        o0 += a * vp[0]; o1 += a * vp[1]; o2 += a * vp[2]; o3 += a * vp[3];
    }
    float* op = out + ((size_t)s * Bn + bn) * 32 + h * 4;
    op[0] = o0; op[1] = o1; op[2] = o2; op[3] = o3;
}

// x[row] = LN(x[row] + r[row]); d = 32 fixed; eps 1e-5
__global__ void k_residual_ln(float* __restrict__ x, const float* __restrict__ r,
                              const float* __restrict__ g, const float* __restrict__ b,
                              long long rows) {
    long long row = (long long)blockIdx.x * blockDim.x + threadIdx.x;
    if (row >= rows) return;
    float* xp = x + row * 32;
    const float* rp = r + row * 32;
    float t[32];
    float mu = 0.f;
    #pragma unroll
    for (int i = 0; i < 32; ++i) { t[i] = xp[i] + rp[i]; mu += t[i]; }
    mu *= (1.f / 32.f);
    float var = 0.f;
    #pragma unroll
    for (int i = 0; i < 32; ++i) { float d = t[i] - mu; var += d * d; }
    var *= (1.f / 32.f);
    float inv = rsqrtf(var + 1e-5f);
    #pragma unroll
    for (int i = 0; i < 32; ++i) xp[i] = g[i] * (t[i] - mu) * inv + b[i];
}

// =================================================================
//  Host orchestration
// =================================================================
static inline int gcount(long long n, int bs) { return (int)((n + bs - 1) / bs); }

extern "C" void kernel_launch(void* const* d_in, const int* in_sizes, int n_in,
                              void* d_out, int out_size, void* d_ws, size_t ws_size,
                              hipStream_t stream) {
    (void)in_sizes; (void)n_in; (void)out_size;
    if (ws_size < WS_NEED * sizeof(float)) return;   // need ~139 MB scratch
    float* ws = (float*)d_ws;

#define FP(i) ((const float*)d_in[(i)])
#define IP(i) ((const int*)d_in[(i)])

    const int*   adj   = IP(0);              // [2, 76800]
    const int*   dei   = IP(1);              // [2, 576000]
    const float* dea   = FP(2);              // [576000, 1]
    const float* tench = FP(3);              // [4, 12, 2]
    const int*   wkday = IP(4);              // [4]
    // d_in[5]=B, d_in[6]=T scalars (constants here)
    // params flat order (insertion order, recursive):
    //  7 node_embed, 8 weekday_embed,
    //  9..32  short (2 enc x 2 conv x 6),  33..80 long (4 enc x 2 conv x 6),
    //  81..122 dyn (3 enc x 2 conv x 7),  123 Wt, 124 bt, 125 pos_learnable,
    //  126..151 dec layer0 (26), 152..177 dec layer1 (26)

    auto gemm = [&](const float* A, const float* B, const float* bias, float* C,
                    int M, int N, int K, int lda, int ldb, int ldc, int transB, int act) {
        int tiles = (M >> 4) * (N >> 4);
        k_wmma_gemm<<<gcount(tiles, 8), 256, 0, stream>>>(A, B, bias, C, M, N, K,
                                                          lda, ldb, ldc, transB, act);
    };
    auto fill = [&](float* p, float v, long long n) {
        k_fill<<<gcount(n, 256), 256, 0, stream>>>(p, v, n);
    };

    // One GATv2 conv: out (pre-zeroed here) = elu(aggregate + bias)
    auto run_conv = [&](const float* x, int ind, int heads, int outd, int outc,
                        const int* esrc, const int* edst, int E, int n,
                        const float* ea, const float* eamean,
                        int pb, bool hasWe, float* out) {
        const float* Wl  = FP(pb + 0); const float* bl  = FP(pb + 1);
        const float* Wr  = FP(pb + 2); const float* br  = FP(pb + 3);
        const float* att = FP(pb + 4); const float* bias = FP(pb + 5);
        const float* We  = hasWe ? FP(pb + 6) : (const float*)0;
        const int hc = heads * outd;
        float* xl = ws + OFF_XL;
        float* xr = ws + OFF_XR;
        gemm(x, Wl, bl, xl, n, hc, ind, ind, hc, hc, 0, 0);
        gemm(x, Wr, br, xr, n, hc, ind, ind, hc, hc, 0, 0);
        fill(ws + OFF_MAXB, 0.f, (long long)n * heads);   // uint 0 == -inf sentinel
        fill(ws + OFF_DEN,  0.f, (long long)n * heads);
        fill(out, 0.f, (long long)n * hc);
        long long totE = (long long)(E + n) * heads;
        k_gat_edge_score<<<gcount(totE, 256), 256, 0, stream>>>(
            esrc, edst, E, n, xl, xr, att, We, ea, eamean,
            ws + OFF_E, (unsigned*)(ws + OFF_MAXB), heads, outd);
        k_gat_finalize_max<<<gcount((long long)n * heads, 256), 256, 0, stream>>>(
            (unsigned*)(ws + OFF_MAXB), ws + OFF_MAXB, n * heads);
        k_gat_expsum<<<gcount(totE, 256), 256, 0, stream>>>(
            edst, E, n, ws + OFF_E, ws + OFF_MAXB, ws + OFF_DEN, heads);
        k_gat_scatter<<<gcount(totE, 256), 256, 0, stream>>>(
            esrc, edst, E, n, ws + OFF_E, ws + OFF_DEN, xl, out, heads, outd);
        k_bias_act<<<gcount((long long)n * outc, 256), 256, 0, stream>>>(
            out, bias, (long long)n, outc, 2 /*elu*/);
    };

    // Encoder stack; result ends in ws+OFF_X
    auto run_stack = [&](const float* x_in, int ind0, int nenc, int base, int stride,
                         const int* esrc, const int* edst, int E, int n,
                         const float* ea, const float* eamean) {
        const float* cur = x_in;
        int ind = ind0;
        for (int e2 = 0; e2 < nenc; ++e2) {
            int pbA = base + (e2 * 2 + 0) * stride;
            int pbB = base + (e2 * 2 + 1) * stride;
            run_conv(cur, ind, 4, 32, 128, esrc, edst, E, n, ea, eamean,
                     pbA, stride == 7, ws + OFF_Y);
            run_conv(ws + OFF_Y, 128, 1, 32, 32, esrc, edst, E, n, ea, eamean,
                     pbB, stride == 7, ws + OFF_X);
            cur = ws + OFF_X;
            ind = 32;
        }
    };

    // ---------- 1. features ----------
    k_build_feat<<<gcount((long long)kBN * 24, 256), 256, 0, stream>>>(
        FP(7), FP(8), wkday, ws + OFF_FEAT);

    // ---------- 2. short / long spatial stacks ----------
    run_stack(ws + OFF_FEAT, 24, 2, 9, 6, adj, adj + kEst, kEst, kBN, 0, 0);
    hipMemcpyAsync(ws + OFF_SHORT, ws + OFF_X, (size_t)kBN * 32 * sizeof(float),
                   hipMemcpyDeviceToDevice, stream);
    run_stack(ws + OFF_FEAT, 24, 4, 33, 6, adj, adj + kEst, kEst, kBN, 0, 0);
    k_add<<<gcount((long long)kBN * 32, 256), 256, 0, stream>>>(
        ws + OFF_X, ws + OFF_SHORT, ws + OFF_SPAT, (long long)kBN * 32);

    // ---------- 3. dyn stack ----------
    fill(ws + OFF_MEAN, 0.f, 1);
    k_reduce_sum<<<256, 256, 0, stream>>>(dea, (long long)kEdy, ws + OFF_MEAN);
    k_scale1<<<1, 32, 0, stream>>>(ws + OFF_MEAN, 1.f / (float)kEdy);
    k_broadcast_dyn<<<gcount((long long)kBTN * 32, 256), 256, 0, stream>>>(
        ws + OFF_SPAT, ws + OFF_X);
    run_stack(ws + OFF_X, 32, 3, 81, 7, dei, dei + kEdy, kEdy, kBTN, dea, ws + OFF_MEAN);

    // ---------- 4. fusion ----------
    k_small_gemm<<<gcount(kB * kT * 32, 256), 256, 0, stream>>>(
        tench, FP(123), FP(124), ws + OFF_TENC, kB * kT, 32, 2);
    k_fused_build<<<gcount((long long)kT * kBN * 32, 256), 256, 0, stream>>>(
        ws + OFF_X, ws + OFF_SPAT, ws + OFF_TENC, FP(125), ws + OFF_FUSED);

    // ---------- 5. transformer decoder ----------
    float* decx  = ws + OFF_DECX;
    float* qb    = ws + OFF_Q;
    float* kb    = ws + OFF_K;
    float* vb    = ws + OFF_V;
    float* attb  = ws + OFF_ATT;
    float* ffh   = ws + OFF_FFH;
    float* proj  = ws + OFF_PROJ;
    float* fused = ws + OFF_FUSED;
    const long long SR = (long long)kFut * kBN;         // 28800 decoder rows
    const long long MR = (long long)kT * kBN;           // 57600 memory rows

    fill(decx, 0.f, SR * 32);
    for (int l = 0; l < 2; ++l) {
        const int pb = 126 + l * 26;
        // self attention
        gemm(decx, FP(pb + 0), FP(pb + 1), qb, (int)SR, 32, 32, 32, 32, 32, 0, 0);
        gemm(decx, FP(pb + 2), FP(pb + 3), kb, (int)SR, 32, 32, 32, 32, 32, 0, 0);
        gemm(decx, FP(pb + 4), FP(pb + 5), vb, (int)SR, 32, 32, 32, 32, 32, 0, 0);
        k_mha_attn<<<gcount((long long)kFut * kBN * 8, 256), 256, 0, stream>>>(
            qb, kb, vb, attb, kFut, kFut, kBN);
        gemm(attb, FP(pb + 6), FP(pb + 7), proj, (int)SR, 32, 32, 32, 32, 32, 0, 0);
        k_residual_ln<<<gcount(SR, 256), 256, 0, stream>>>(
            decx, proj, FP(pb + 20), FP(pb + 21), SR);
        // cross attention
        gemm(decx,  FP(pb + 8),  FP(pb + 9),  qb, (int)SR, 32, 32, 32, 32, 32, 0, 0);
        gemm(fused, FP(pb + 10), FP(pb + 11), kb, (int)MR, 32, 32, 32, 32, 32, 0, 0);
        gemm(fused, FP(pb + 12), FP(pb + 13), vb, (int)MR, 32, 32, 32, 32, 32, 0, 0);
        k_mha_attn<<<gcount((long long)kFut * kBN * 8, 256), 256, 0, stream>>>(
            qb, kb, vb, attb, kFut, kT, kBN);
        gemm(attb, FP(pb + 14), FP(pb + 15), proj, (int)SR, 32, 32, 32, 32, 32, 0, 0);
        k_residual_ln<<<gcount(SR, 256), 256, 0, stream>>>(
            decx, proj, FP(pb + 22), FP(pb + 23), SR);
        // feed-forward
        gemm(decx, FP(pb + 16), FP(pb + 17), ffh, (int)SR, 128, 32, 32, 128, 128, 0, 1 /*relu*/);
        gemm(ffh,  FP(pb + 18), FP(pb + 19), proj, (int)SR, 32, 128, 128, 32, 32, 0, 0);
        k_residual_ln<<<gcount(SR, 256), 256, 0, stream>>>(
            decx, proj, FP(pb + 24), FP(pb + 25), SR);
    }

    // ---------- 6. final einsum: out[b,f,:,:] = D D^T (WMMA A@B^T, K=32) ----------
    float* fout = (float*)d_out;
    for (int b = 0; b < kB; ++b)
        for (int f = 0; f < kFut; ++f) {
            const float* Am = decx + ((size_t)f * kBN + (size_t)b * kN) * 32;
            float* Co = fout + ((size_t)b * kFut + f) * kN * kN;
            gemm(Am, Am, (const float*)0, Co, kN, kN, 32, 32, 32, kN, 1, 0);
        }
#undef FP
#undef IP
}